// BayesianSkipGramModel_70489003262718
// MI455X (gfx1250) — compile-verified
//
#include <hip/hip_runtime.h>
#include <hip/hip_bf16.h>
#include <math.h>

#define V_SZ 50000
#define D_SZ 300
#define Z_SZ 128
#define C_SZ 10
#define NTILE ((V_SZ + 15) / 16)        // 3125 row-tiles of 16
#define WPB 8                           // waves per block in GEMV kernel
#define GEMV_BLOCKS ((NTILE + WPB - 1) / WPB)

typedef __attribute__((ext_vector_type(2))) float v2f;
typedef __attribute__((ext_vector_type(8))) float v8f;

__device__ __forceinline__ float softplusf(float x) {
    // numerically stable log(1 + exp(x))
    return fmaxf(x, 0.0f) + log1pf(expf(-fabsf(x)));
}

// ---------------------------------------------------------------------------
// Kernel 1: encoder side (tiny). One block, 256 threads.
//   - locate center index from one-hot
//   - summed[600] = [C*relu(E[:,ci]) ; sum_c relu(E[:,idx_c])]
//   - u = W_mu@summed + b_mu ; s = softplus(W_sig@summed + b_sig)
//   - z = u + eps*s ; z_sigma = softplus(prior_sigma[:,ci]) ; KL sum
// ---------------------------------------------------------------------------
__global__ void bsg_encode_kernel(const float* __restrict__ center_word,
                                  const int*   __restrict__ idxs,
                                  const float* __restrict__ eps,
                                  const float* __restrict__ E,
                                  const float* __restrict__ W_mu,
                                  const float* __restrict__ b_mu,
                                  const float* __restrict__ W_sig,
                                  const float* __restrict__ b_sig,
                                  const float* __restrict__ prior_sigma,
                                  float* __restrict__ zvec_out,
                                  float* __restrict__ klsum_out) {
    __shared__ int   s_cidx;
    __shared__ float s_summed[2 * D_SZ];
    __shared__ float s_red[256];
    const int t = threadIdx.x;

    // find the one-hot center index
    for (int i = t; i < V_SZ; i += 256)
        if (center_word[i] > 0.5f) s_cidx = i;
    __syncthreads();
    const int cidx = s_cidx;

    // summed[0:D]  = C * relu(ce) ; summed[D:2D] = sum_c relu(cw_c)
    for (int d = t; d < D_SZ; d += 256) {
        const float* Erow = E + (size_t)d * V_SZ;
        s_summed[d] = (float)C_SZ * fmaxf(Erow[cidx], 0.0f);
        float acc = 0.0f;
        #pragma unroll
        for (int c = 0; c < C_SZ; ++c)
            acc += fmaxf(Erow[idxs[c]], 0.0f);
        s_summed[D_SZ + d] = acc;
    }
    __syncthreads();

    float kl = 0.0f;
    if (t < Z_SZ) {
        const float* wm = W_mu  + (size_t)t * (2 * D_SZ);
        const float* wg = W_sig + (size_t)t * (2 * D_SZ);
        float um = b_mu[t];
        float us = b_sig[t];
        for (int j = 0; j < 2 * D_SZ; ++j) {
            const float sj = s_summed[j];
            um = fmaf(wm[j], sj, um);
            us = fmaf(wg[j], sj, us);
        }
        const float sgm = softplusf(us);
        zvec_out[t] = fmaf(eps[t], sgm, um);             // z = u + eps*s
        const float zsig = softplusf(prior_sigma[(size_t)t * V_SZ + cidx]);
        const float du = um - zsig;
        kl = logf(zsig / sgm) + (sgm * sgm + du * du) / (2.0f * zsig * zsig) - 0.5f;
    }

    // block reduction of KL
    s_red[t] = kl;
    __syncthreads();
    for (int off = 128; off > 0; off >>= 1) {
        if (t < off) s_red[t] += s_red[t + off];
        __syncthreads();
    }
    if (t == 0) *klsum_out = s_red[0];
}

// ---------------------------------------------------------------------------
// Kernel 2: logits = W_gen @ z + b_gen, via V_WMMA_F32_16X16X4_F32.
// One wave per 16-row tile of W_gen. A = W_gen tile (documented 16x4 f32
// layout: lanes 0-15 hold K=0,1; lanes 16-31 hold K=2,3 of rows M=lane&15).
// B broadcasts z[k] across all 16 columns (value depends only on K), so all
// columns of D carry the same 16 dot products; 32 chained WMMAs cover K=128.
// ---------------------------------------------------------------------------
__global__ void bsg_gemv_wmma_kernel(const float* __restrict__ W_gen,
                                     const float* __restrict__ b_gen,
                                     const float* __restrict__ zvec,
                                     float* __restrict__ logits) {
    __shared__ float zs[Z_SZ];
    const int t = threadIdx.x;
    if (t < Z_SZ) zs[t] = zvec[t];
    __syncthreads();

    const int wave = t >> 5;
    const int lane = t & 31;
    const int tile = blockIdx.x * WPB + wave;
    if (tile >= NTILE) return;                 // uniform per wave; EXEC stays full

    const int row   = tile * 16 + (lane & 15); // M for this lane
    const int khalf = (lane >> 4) * 2;         // lanes 0-15 -> K 0,1 ; 16-31 -> K 2,3
    const float* __restrict__ wrow = W_gen + (size_t)row * Z_SZ + khalf;
    const float* __restrict__ zrow = zs + khalf;

    v8f acc = {0.f, 0.f, 0.f, 0.f, 0.f, 0.f, 0.f, 0.f};
    #pragma unroll 4
    for (int k0 = 0; k0 < Z_SZ; k0 += 4) {
        v2f a; a.x = wrow[k0];     a.y = wrow[k0 + 1];
        v2f b; b.x = zrow[k0];     b.y = zrow[k0 + 1];
        acc = __builtin_amdgcn_wmma_f32_16x16x4_f32(
            /*neg_a=*/false, a, /*neg_b=*/false, b,
            /*c_mod=*/(short)0, acc, /*reuse_a=*/false, /*reuse_b=*/false);
    }

    // D layout: VGPR j holds row M=j (lanes 0-15) / M=j+8 (lanes 16-31);
    // every column is identical, so lanes 0 and 16 emit the 16 row results.
    if ((lane & 15) == 0) {
        const int mbase = tile * 16 + (lane >> 4) * 8;
        #pragma unroll
        for (int j = 0; j < 8; ++j)
            logits[mbase + j] = acc[j] + b_gen[mbase + j];
    }
}

// ---------------------------------------------------------------------------
// Kernel 3: log-softmax reduction + gather + final scalar. One block, 1024 thr.
// ---------------------------------------------------------------------------
__global__ void bsg_reduce_kernel(const float* __restrict__ logits,
                                  const int*   __restrict__ idxs,
                                  const float* __restrict__ klsum,
                                  float* __restrict__ out) {
    __shared__ float red[1024];
    const int t = threadIdx.x;

    float m = -INFINITY;
    for (int i = t; i < V_SZ; i += 1024) m = fmaxf(m, logits[i]);
    red[t] = m;
    __syncthreads();
    for (int off = 512; off > 0; off >>= 1) {
        if (t < off) red[t] = fmaxf(red[t], red[t + off]);
        __syncthreads();
    }
    m = red[0];
    __syncthreads();

    float sum = 0.0f;
    for (int i = t; i < V_SZ; i += 1024) sum += expf(logits[i] - m);
    red[t] = sum;
    __syncthreads();
    for (int off = 512; off > 0; off >>= 1) {
        if (t < off) red[t] += red[t + off];
        __syncthreads();
    }

    if (t == 0) {
        const float lse = m + logf(red[0]);
        float lp = 0.0f;
        #pragma unroll
        for (int c = 0; c < C_SZ; ++c)
            lp += logits[idxs[c]] - lse;
        out[0] = lp - klsum[0];
    }
}

extern "C" void kernel_launch(void* const* d_in, const int* in_sizes, int n_in,
                              void* d_out, int out_size, void* d_ws, size_t ws_size,
                              hipStream_t stream) {
    const float* center_word = (const float*)d_in[0];
    // d_in[1] context_words (one-hot) unused: idxs give identical gathers
    const int*   idxs        = (const int*)  d_in[2];
    const float* eps         = (const float*)d_in[3];
    const float* E           = (const float*)d_in[4];
    const float* W_mu        = (const float*)d_in[5];
    const float* b_mu        = (const float*)d_in[6];
    const float* W_sig       = (const float*)d_in[7];
    const float* b_sig       = (const float*)d_in[8];
    // d_in[9] prior_mean unused by the reference loss
    const float* prior_sigma = (const float*)d_in[10];
    const float* W_gen       = (const float*)d_in[11];
    const float* b_gen       = (const float*)d_in[12];
    float* out = (float*)d_out;

    float* ws     = (float*)d_ws;
    float* zvec   = ws;            // 128 floats
    float* klsum  = ws + 128;      // 1 float
    float* logits = ws + 160;      // 50000 floats (16B-aligned offset)

    bsg_encode_kernel<<<1, 256, 0, stream>>>(center_word, idxs, eps, E, W_mu, b_mu,
                                             W_sig, b_sig, prior_sigma, zvec, klsum);
    bsg_gemv_wmma_kernel<<<GEMV_BLOCKS, WPB * 32, 0, stream>>>(W_gen, b_gen, zvec, logits);
    bsg_reduce_kernel<<<1, 1024, 0, stream>>>(logits, idxs, klsum, out);
}